// SparseMixerV2_67671504716019
// MI455X (gfx1250) — compile-verified
//
#include <hip/hip_runtime.h>
#include <stdint.h>

// ---------------------------------------------------------------------------
// SparseMixer top-1 forward, MI455X (gfx1250, wave32).
//
// Per row (E=64): sel = first argmax(logits); multiplier = 1/sum_kept exp(l-mlt)
// (mask_for_one is provably 1.0: the unmasked global max is always the masked
//  softmax argmax, so sel==max_ind; rand_u is dead in the forward output).
//
// Bandwidth-bound: 268 MB read-once @ 23.3 TB/s ~= 11.5 us floor. Data is
// streamed through LDS with the CDNA5 async copy engine
// (global_load_async_to_lds_b128 + ASYNCcnt), triple-buffered so each wave
// keeps two 2KB tiles (8 async B128 ops) in flight ahead of compute.
// Loads are tagged th:TH_LOAD_NT: the stream exceeds the 192MB L2 and has
// zero reuse, so non-temporal is the correct cache policy.
// ---------------------------------------------------------------------------

#define WAVES_PER_BLOCK 8
#define ROWS_PER_TILE   8                        // rows per wave per tile
#define TILE_BYTES      (ROWS_PER_TILE * 64 * 4) // 2048 B = 4 x (32 lanes x 16B)
#define NBUF            3                        // LDS ring depth (triple buffer)
#define LDS_BYTES_PER_WAVE (NBUF * TILE_BYTES)   // 6 KB/wave -> 48 KB/block

__device__ __forceinline__ void async_wait_le8() {
    asm volatile("s_wait_asynccnt 0x8" ::: "memory");
}
__device__ __forceinline__ void async_wait_le4() {
    asm volatile("s_wait_asynccnt 0x4" ::: "memory");
}
__device__ __forceinline__ void async_wait_le0() {
    asm volatile("s_wait_asynccnt 0x0" ::: "memory");
}

// Issue one 2KB tile: 4 async B128 copies, 16B per lane per copy, linear.
__device__ __forceinline__ void issue_tile_async(const float* __restrict__ logits,
                                                 uint64_t lastValidByte,
                                                 uint64_t tileByteBase,
                                                 uint32_t ldsBase, int lane) {
#pragma unroll
    for (int j = 0; j < 4; ++j) {
        uint32_t off = (uint32_t)(j * 512 + lane * 16);
        uint64_t g = (uint64_t)(uintptr_t)logits + tileByteBase + off;
        if (g > lastValidByte) g = lastValidByte;   // tail clamp (dup src is fine)
        uint32_t l = ldsBase + off;
        asm volatile("global_load_async_to_lds_b128 %0, %1, off th:TH_LOAD_NT"
                     :: "v"(l), "v"(g) : "memory");
    }
}

__global__ void __launch_bounds__(256)
sparsemixer_top1_fwd(const float* __restrict__ logits,
                     float* __restrict__ out_sel,
                     float* __restrict__ out_mult,
                     float* __restrict__ out_loss,
                     int N) {
    __shared__ __align__(16) float lds[WAVES_PER_BLOCK * (LDS_BYTES_PER_WAVE / 4)];

    const int tid       = (int)threadIdx.x;
    const int lane      = tid & 31;
    const int waveInBlk = tid >> 5;
    const int waveGlb   = (int)blockIdx.x * WAVES_PER_BLOCK + waveInBlk;
    const int nWaves    = (int)gridDim.x * WAVES_PER_BLOCK;
    const int numTiles  = (N + ROWS_PER_TILE - 1) / ROWS_PER_TILE;

    if (tid == 0 && blockIdx.x == 0) out_loss[0] = 0.0f;   // balance_loss

    float* myLds = &lds[waveInBlk * (LDS_BYTES_PER_WAVE / 4)];
    const uint32_t ldsBase0 = (uint32_t)(uintptr_t)myLds;  // low 32 bits = LDS offset
    const uint64_t lastValidByte =
        (uint64_t)(uintptr_t)logits + (uint64_t)N * 256u - 16u;

    const int tile0 = waveGlb;
    if (tile0 >= numTiles) return;

    // --- pipeline prologue: prime two tiles into the LDS ring
    issue_tile_async(logits, lastValidByte, (uint64_t)tile0 * TILE_BYTES,
                     ldsBase0, lane);
    if (tile0 + nWaves < numTiles) {
        issue_tile_async(logits, lastValidByte,
                         (uint64_t)(tile0 + nWaves) * TILE_BYTES,
                         ldsBase0 + TILE_BYTES, lane);
    }

    const int half = lane >> 4;   // 0: lanes 0-15, 1: lanes 16-31
    const int h    = lane & 15;
    int bufIdx = 0;

    for (int tile = tile0; tile < numTiles; tile += nWaves) {
        // --- keep two tiles in flight; exact waits via in-order ASYNCcnt
        const int t2 = tile + 2 * nWaves;
        if (t2 < numTiles) {
            int b2 = bufIdx + 2; if (b2 >= NBUF) b2 -= NBUF;
            issue_tile_async(logits, lastValidByte, (uint64_t)t2 * TILE_BYTES,
                             ldsBase0 + (uint32_t)b2 * TILE_BYTES, lane);
            async_wait_le8();        // 12 outstanding -> <=8 means current tile done
        } else if (tile + nWaves < numTiles) {
            async_wait_le4();        // 8 outstanding  -> <=4 means current tile done
        } else {
            async_wait_le0();
        }

        const float* tbuf = myLds + (size_t)bufIdx * (TILE_BYTES / 4);

#pragma unroll
        for (int p = 0; p < 4; ++p) {
            const int r = 2 * p + half;                       // row-in-tile for this half
            const long long row = (long long)tile * ROWS_PER_TILE + r;
            const float4 v = *(const float4*)&tbuf[r * 64 + h * 4];  // ds_load_b128

            // ---- max + first-index within this lane's 4 elements
            float m = v.x; int idx = h * 4;
            if (v.y > m) { m = v.y; idx = h * 4 + 1; }
            if (v.z > m) { m = v.z; idx = h * 4 + 2; }
            if (v.w > m) { m = v.w; idx = h * 4 + 3; }

            // ---- 16-lane butterfly (masks < 16 stay inside the half; wave32-safe)
#pragma unroll
            for (int d = 1; d < 16; d <<= 1) {
                float om = __shfl_xor(m, d, 32);
                int   oi = __shfl_xor(idx, d, 32);
                if (om > m || (om == m && oi < idx)) { m = om; idx = oi; }
            }

            // ---- masked exp-sum: keep iff (mlt-l)/max(|l|,mlt) <= 0.2
            // (division-free; factor==0 => 0<=0 keep, matching NaN>0.2==false)
            float s = 0.0f;
            {
                float fa = fmaxf(fabsf(v.x), m); if ((m - v.x) <= 0.2f * fa) s += __expf(v.x - m);
                float fb = fmaxf(fabsf(v.y), m); if ((m - v.y) <= 0.2f * fb) s += __expf(v.y - m);
                float fc = fmaxf(fabsf(v.z), m); if ((m - v.z) <= 0.2f * fc) s += __expf(v.z - m);
                float fd = fmaxf(fabsf(v.w), m); if ((m - v.w) <= 0.2f * fd) s += __expf(v.w - m);
            }
#pragma unroll
            for (int d = 1; d < 16; d <<= 1) s += __shfl_xor(s, d, 32);

            // ---- write (sel always equals masked argmax => mask_for_one == 1.0)
            if (h == 0 && row < (long long)N) {
                out_sel[row]  = (float)idx;
                out_mult[row] = 1.0f / s;   // = masked_gates[sel] * 1.0
            }
        }

        ++bufIdx; if (bufIdx >= NBUF) bufIdx = 0;
    }
}

extern "C" void kernel_launch(void* const* d_in, const int* in_sizes, int n_in,
                              void* d_out, int out_size, void* d_ws, size_t ws_size,
                              hipStream_t stream) {
    (void)n_in; (void)d_ws; (void)ws_size; (void)out_size;
    const float* logits = (const float*)d_in[0];
    // d_in[1] (rand_u) is provably dead in the forward output; never read.

    const int NE = in_sizes[0];     // N * 64
    const int N  = NE / 64;

    float* out      = (float*)d_out;
    float* out_sel  = out;                   // [0, N)  : sel (as float values)
    float* out_mult = out + N;               // [N, 2N) : multiplier
    float* out_loss = out + 2 * (size_t)N;   // [2N]    : balance_loss = 0

    const int blocks = 2048;                 // 16384 waves; 8 tiles/wave at N=1M
    hipLaunchKernelGGL(sparsemixer_top1_fwd, dim3(blocks), dim3(256), 0, stream,
                       logits, out_sel, out_mult, out_loss, N);
}